// MSAlphaCrossAttn_7447473291692
// MI455X (gfx1250) — compile-verified
//
#include <hip/hip_runtime.h>

typedef __attribute__((ext_vector_type(2))) float v2f;
typedef __attribute__((ext_vector_type(8))) float v8f;

#define N_BATCH 4
#define LQ      1000
#define DM      256
#define S_TOT   19822
#define IMG_F   976.0f

__constant__ int LVL_H[4]     = {122, 61, 31, 16};
__constant__ int LVL_W[4]     = {122, 61, 31, 16};
__constant__ int LVL_START[4] = {0, 14884, 18605, 19566};

#define WMMA_F32(A, B, C) __builtin_amdgcn_wmma_f32_16x16x4_f32( \
    false, (A), false, (B), (short)0, (C), false, false)

// ---------------------------------------------------------------------------
// f32 GEMM: Y[b^xor][row][col] = X[b][row][:] . W[col][:] + bias[col]
// One wave32 computes a 32x64 tile (2 row-frags x 4 col-frags).
// Software-pipelined: the next k-step's 6 b64 loads are issued before the
// current k-step's 8 V_WMMA_F32_16X16X4_F32, so matrix ops overlap memory.
// A-frag (ISA 16x4): lanes 0-15 -> K=k..k+1, lanes 16-31 -> K=k+2..k+3,
// row = lane&15. B-frag mirrors with N = lane&15. D: M = r + 8*(lane>>4).
// ---------------------------------------------------------------------------
__global__ __launch_bounds__(256) void gemm_wmma_f32(
    const float* __restrict__ X,    // [N_BATCH][rows][K]
    const float* __restrict__ W,    // [cols][K] row-major (multiply by W^T)
    const float* __restrict__ bias, // [cols]
    float* __restrict__ Y,          // [N_BATCH][rows][cols]
    int rows, int K, int cols, int batchXor)
{
    const int lane = threadIdx.x & 31;
    const int wave = (int)((blockIdx.x * blockDim.x + threadIdx.x) >> 5);
    const int colTiles = cols >> 6;          // 64-wide wave tiles
    const int rowTiles = (rows + 31) >> 5;   // 32-high wave tiles
    const int tilesPerBatch = rowTiles * colTiles;
    if (wave >= tilesPerBatch * N_BATCH) return;

    int b = wave / tilesPerBatch;
    int t = wave - b * tilesPerBatch;
    int rt = t / colTiles;
    int ct = t - rt * colTiles;
    const int row0 = rt << 5, col0 = ct << 6;

    const int half = lane >> 4;              // 0: K+0..1, 1: K+2..3
    const int l    = lane & 15;

    int ar0 = row0 + l;      if (ar0 > rows - 1) ar0 = rows - 1;
    int ar1 = row0 + 16 + l; if (ar1 > rows - 1) ar1 = rows - 1;

    // Independent base pointers so each k-step's loads clause together.
    const float* a0p = X + ((long)b * rows + ar0) * K + (half << 1);
    const float* a1p = X + ((long)b * rows + ar1) * K + (half << 1);
    const float* bp0 = W + (long)(col0 + l)      * K + (half << 1);
    const float* bp1 = bp0 + (long)16 * K;
    const float* bp2 = bp0 + (long)32 * K;
    const float* bp3 = bp0 + (long)48 * K;

    v8f acc[2][4];
    #pragma unroll
    for (int i = 0; i < 2; ++i)
        #pragma unroll
        for (int c = 0; c < 4; ++c)
            acc[i][c] = (v8f){0.f,0.f,0.f,0.f,0.f,0.f,0.f,0.f};

    // Prologue: k = 0 fragments
    v2f a0 = *(const v2f*)(a0p);
    v2f a1 = *(const v2f*)(a1p);
    v2f b0 = *(const v2f*)(bp0);
    v2f b1 = *(const v2f*)(bp1);
    v2f b2 = *(const v2f*)(bp2);
    v2f b3 = *(const v2f*)(bp3);

    for (int k = 4; k < K; k += 4) {
        // Issue next k-step's loads first (immediate-offset, clauseable)
        v2f na0 = *(const v2f*)(a0p + k);
        v2f na1 = *(const v2f*)(a1p + k);
        v2f nb0 = *(const v2f*)(bp0 + k);
        v2f nb1 = *(const v2f*)(bp1 + k);
        v2f nb2 = *(const v2f*)(bp2 + k);
        v2f nb3 = *(const v2f*)(bp3 + k);
        // Compute current k-step while next loads are in flight
        acc[0][0] = WMMA_F32(a0, b0, acc[0][0]);
        acc[1][0] = WMMA_F32(a1, b0, acc[1][0]);
        acc[0][1] = WMMA_F32(a0, b1, acc[0][1]);
        acc[1][1] = WMMA_F32(a1, b1, acc[1][1]);
        acc[0][2] = WMMA_F32(a0, b2, acc[0][2]);
        acc[1][2] = WMMA_F32(a1, b2, acc[1][2]);
        acc[0][3] = WMMA_F32(a0, b3, acc[0][3]);
        acc[1][3] = WMMA_F32(a1, b3, acc[1][3]);
        a0 = na0; a1 = na1;
        b0 = nb0; b1 = nb1; b2 = nb2; b3 = nb3;
    }
    // Epilogue: last k-step
    acc[0][0] = WMMA_F32(a0, b0, acc[0][0]);
    acc[1][0] = WMMA_F32(a1, b0, acc[1][0]);
    acc[0][1] = WMMA_F32(a0, b1, acc[0][1]);
    acc[1][1] = WMMA_F32(a1, b1, acc[1][1]);
    acc[0][2] = WMMA_F32(a0, b2, acc[0][2]);
    acc[1][2] = WMMA_F32(a1, b2, acc[1][2]);
    acc[0][3] = WMMA_F32(a0, b3, acc[0][3]);
    acc[1][3] = WMMA_F32(a1, b3, acc[1][3]);

    const int ob = b ^ batchXor;             // pre-apply swap_pairs for value
    #pragma unroll
    for (int c = 0; c < 4; ++c) {
        const int col = col0 + (c << 4) + l;
        const float bs = bias[col];
        float* out = Y + (long)ob * rows * cols + col;
        #pragma unroll
        for (int i = 0; i < 2; ++i) {
            #pragma unroll
            for (int r = 0; r < 8; ++r) {
                int row = row0 + (i << 4) + r + (half << 3);
                if (row < rows) out[(long)row * cols] = acc[i][c][r] + bs;
            }
        }
    }
}

// ---------------------------------------------------------------------------
// Softmax over contiguous rows of 64 (one wave32 per row, 2 elems/lane).
// ---------------------------------------------------------------------------
__global__ __launch_bounds__(256) void softmax64(float* __restrict__ attn, int nRows)
{
    int lane = threadIdx.x & 31;
    int wave = threadIdx.x >> 5;
    int row = blockIdx.x * 8 + wave;
    if (row >= nRows) return;
    float* p = attn + (long)row * 64;
    float a = p[lane], b = p[lane + 32];
    float m = fmaxf(a, b);
    #pragma unroll
    for (int off = 16; off >= 1; off >>= 1) m = fmaxf(m, __shfl_xor(m, off, 32));
    float ea = __expf(a - m), eb = __expf(b - m);
    float s = ea + eb;
    #pragma unroll
    for (int off = 16; off >= 1; off >>= 1) s += __shfl_xor(s, off, 32);
    float inv = 1.0f / s;
    p[lane] = ea * inv;
    p[lane + 32] = eb * inv;
}

// ---------------------------------------------------------------------------
// Geometry: one thread per (n,q). Computes the 16 unique clipped normalized
// points (identical across the 4 levels and 8 heads).
// ---------------------------------------------------------------------------
__global__ __launch_bounds__(256) void geom_kernel(
    const float* __restrict__ refpts, // [4][1000][2]
    const float* __restrict__ pfw,    // [4][3][4]
    const float* __restrict__ pt,     // [4][2][3]
    float* __restrict__ loc)          // [4][1000][16][2]
{
    int idx = blockIdx.x * blockDim.x + threadIdx.x;
    if (idx >= N_BATCH * LQ) return;
    int n = idx / LQ;

    // orig = inv(A2x2) @ (ref - t) * IMG     (unswapped pt_transforms)
    const float* T = pt + n * 6;
    float rx = refpts[idx * 2]     - T[2];
    float ry = refpts[idx * 2 + 1] - T[5];
    float invdet = 1.0f / (T[0] * T[4] - T[1] * T[3]);
    float ox = ( T[4] * rx - T[1] * ry) * invdet * IMG_F;
    float oy = (-T[3] * rx + T[0] * ry) * invdet * IMG_F;

    // 3x3 inverse of A1 = p_pfws[n][:, :3] via adjugate
    const float* P1 = pfw + n * 12;
    float A00=P1[0], A01=P1[1], A02=P1[2],  b0=P1[3];
    float A10=P1[4], A11=P1[5], A12=P1[6],  b1=P1[7];
    float A20=P1[8], A21=P1[9], A22=P1[10], b2=P1[11];
    float c00 =  (A11*A22 - A12*A21);
    float c01 = -(A10*A22 - A12*A20);
    float c02 =  (A10*A21 - A11*A20);
    float id = 1.0f / (A00*c00 + A01*c01 + A02*c02);
    float i00=c00*id, i01=-(A01*A22 - A02*A21)*id, i02= (A01*A12 - A02*A11)*id;
    float i10=c01*id, i11= (A00*A22 - A02*A20)*id, i12=-(A00*A12 - A02*A10)*id;
    float i20=c02*id, i21=-(A00*A21 - A01*A20)*id, i22= (A00*A11 - A01*A10)*id;

    // cam = -inv(A1) @ b1 ; ray = inv(A1) @ ([orig,1] - b1)
    float camx = -(i00*b0 + i01*b1 + i02*b2);
    float camy = -(i10*b0 + i11*b1 + i12*b2);
    float camz = -(i20*b0 + i21*b1 + i22*b2);
    float u0 = ox - b0, u1 = oy - b1, u2 = 1.0f - b2;
    float rxx = i00*u0 + i01*u1 + i02*u2;
    float ryy = i10*u0 + i11*u1 + i12*u2;
    float rzz = i20*u0 + i21*u1 + i22*u2;

    // proj2 uses the SWAPPED projection P2 = p_pfws[n^1]
    const float* P2 = pfw + (n ^ 1) * 12;
    float y0, y1, y2, inv2;
    y0 = P2[0]*camx + P2[1]*camy + P2[2]*camz + P2[3];
    y1 = P2[4]*camx + P2[5]*camy + P2[6]*camz + P2[7];
    y2 = P2[8]*camx + P2[9]*camy + P2[10]*camz + P2[11];
    inv2 = 1.0f / (y2 + 1e-8f);
    float tsx = y0 * inv2, tsy = y1 * inv2;

    y0 = P2[0]*rxx + P2[1]*ryy + P2[2]*rzz + P2[3];
    y1 = P2[4]*rxx + P2[5]*ryy + P2[6]*rzz + P2[7];
    y2 = P2[8]*rxx + P2[9]*ryy + P2[10]*rzz + P2[11];
    inv2 = 1.0f / (y2 + 1e-8f);
    float msx = (y0 * inv2 - tsx) * (1.0f / IMG_F);
    float msy = (y1 * inv2 - tsy) * (1.0f / IMG_F);

    // final affine with SWAPPED pt_transforms, clip to [0,1]
    const float* T2 = pt + (n ^ 1) * 6;
    float* L = loc + (long)idx * 32;
    #pragma unroll
    for (int j = 0; j < 16; ++j) {
        float xs = IMG_F * (float)j * (1.0f / 15.0f);   // linspace(0, IMG, 16)
        float lx = (tsx + msx * xs) * (1.0f / IMG_F);
        float ly = (tsy + msy * xs) * (1.0f / IMG_F);
        float fx = T2[0]*lx + T2[1]*ly + T2[2];
        float fy = T2[3]*lx + T2[4]*ly + T2[5];
        fx = fminf(fmaxf(fx, 0.0f), 1.0f);
        fy = fminf(fmaxf(fy, 0.0f), 1.0f);
        L[j * 2]     = fx;
        L[j * 2 + 1] = fy;
    }
}

// ---------------------------------------------------------------------------
// Deformable sampling: one 256-thread block per (n,q); thread = head*32 + ch.
// Each corner tap reads 256 contiguous floats across the block (L2-resident).
// ---------------------------------------------------------------------------
__global__ __launch_bounds__(256) void sample_kernel(
    const float* __restrict__ value, // [4][S][256], batch already swapped
    const float* __restrict__ attn,  // [4][1000][8*64]
    const float* __restrict__ loc,   // [4][1000][16][2]
    float* __restrict__ O)           // [4][1000][256]
{
    const int nq = blockIdx.x;                 // 0 .. 4*LQ-1
    const int n  = nq / LQ;
    const int tid = threadIdx.x;               // m*32 + d
    const int m = tid >> 5;

    const float* vbase = value + (long)n * S_TOT * DM + tid;
    const float* aro   = attn + (long)nq * 512 + m * 64;
    const float* L     = loc + (long)nq * 32;

    float acc = 0.0f;
    for (int lid = 0; lid < 4; ++lid) {
        const int H  = LVL_H[lid];
        const int W  = LVL_W[lid];
        const float* vlvl = vbase + (long)LVL_START[lid] * DM;
        #pragma unroll 4
        for (int j = 0; j < 16; ++j) {
            float w  = aro[lid * 16 + j];
            float gx = L[j * 2]     * (float)W - 0.5f;
            float gy = L[j * 2 + 1] * (float)H - 0.5f;
            float fx0 = floorf(gx), fy0 = floorf(gy);
            float wx = gx - fx0, wy = gy - fy0;
            int x0 = (int)fx0, y0 = (int)fy0;

            float v00, v01, v10, v11;
            {
                int x1 = x0 + 1, y1 = y0 + 1;
                int x0c = x0 < 0 ? 0 : (x0 > W - 1 ? W - 1 : x0);
                int x1c = x1 < 0 ? 0 : (x1 > W - 1 ? W - 1 : x1);
                int y0c = y0 < 0 ? 0 : (y0 > H - 1 ? H - 1 : y0);
                int y1c = y1 < 0 ? 0 : (y1 > H - 1 ? H - 1 : y1);
                bool vx0 = (x0 >= 0) & (x0 < W);
                bool vx1 = (x1 >= 0) & (x1 < W);
                bool vy0 = (y0 >= 0) & (y0 < H);
                bool vy1 = (y1 >= 0) & (y1 < H);
                v00 = (vx0 & vy0) ? vlvl[(long)(y0c * W + x0c) * DM] : 0.0f;
                v01 = (vx1 & vy0) ? vlvl[(long)(y0c * W + x1c) * DM] : 0.0f;
                v10 = (vx0 & vy1) ? vlvl[(long)(y1c * W + x0c) * DM] : 0.0f;
                v11 = (vx1 & vy1) ? vlvl[(long)(y1c * W + x1c) * DM] : 0.0f;
            }
            float bil = v00 * (1.0f - wx) * (1.0f - wy)
                      + v01 * wx * (1.0f - wy)
                      + v10 * (1.0f - wx) * wy
                      + v11 * wx * wy;
            acc = fmaf(w, bil, acc);
        }
    }
    O[(long)nq * DM + tid] = acc;
}

// ---------------------------------------------------------------------------
// Workspace layout (all 256B aligned), total ~94 MB
// ---------------------------------------------------------------------------
static const size_t OFF_VALUE = 0;                                  // 4*19822*256 f32
static const size_t OFF_ATTN  = (size_t)4 * S_TOT * DM * 4;         // 81,190,912
static const size_t OFF_LOC   = OFF_ATTN + (size_t)4 * LQ * 512 * 4;
static const size_t OFF_O     = OFF_LOC + (size_t)4 * LQ * 32 * 4;

static inline int gemm_blocks(int rows, int cols) {
    int tiles = ((rows + 31) >> 5) * (cols >> 6) * N_BATCH;   // waves needed
    return (tiles + 7) / 8;                                   // 8 waves / block
}

extern "C" void kernel_launch(void* const* d_in, const int* in_sizes, int n_in,
                              void* d_out, int out_size, void* d_ws, size_t ws_size,
                              hipStream_t stream) {
    const float* query         = (const float*)d_in[0];
    const float* input_flatten = (const float*)d_in[1];
    const float* refp          = (const float*)d_in[2];
    const float* p_pfws        = (const float*)d_in[3];
    const float* ptt           = (const float*)d_in[4];
    const float* W_val         = (const float*)d_in[5];
    const float* b_val         = (const float*)d_in[6];
    const float* W_attn        = (const float*)d_in[7];
    const float* b_attn        = (const float*)d_in[8];
    const float* W_out         = (const float*)d_in[9];
    const float* b_out         = (const float*)d_in[10];
    float* out = (float*)d_out;

    char* ws = (char*)d_ws;
    float* value_ws = (float*)(ws + OFF_VALUE);
    float* attn_ws  = (float*)(ws + OFF_ATTN);
    float* loc_ws   = (float*)(ws + OFF_LOC);
    float* O_ws     = (float*)(ws + OFF_O);

    // 1) value = input_flatten @ W_val^T + b_val, written batch-swapped (n^1)
    gemm_wmma_f32<<<gemm_blocks(S_TOT, DM), 256, 0, stream>>>(
        input_flatten, W_val, b_val, value_ws, S_TOT, DM, DM, 1);
    // 2) attn logits = query @ W_attn^T + b_attn   -> [4][1000][512]
    gemm_wmma_f32<<<gemm_blocks(LQ, 512), 256, 0, stream>>>(
        query, W_attn, b_attn, attn_ws, LQ, DM, 512, 0);
    // 3) softmax over the 64 points of each (n,q,head)
    {
        int rows = N_BATCH * LQ * 8;
        softmax64<<<(rows + 7) / 8, 256, 0, stream>>>(attn_ws, rows);
    }
    // 4) sampling-location geometry
    geom_kernel<<<(N_BATCH * LQ + 255) / 256, 256, 0, stream>>>(
        refp, p_pfws, ptt, loc_ws);
    // 5) bilinear gather + weighted sum  -> O [4][1000][256]
    sample_kernel<<<N_BATCH * LQ, 256, 0, stream>>>(value_ws, attn_ws, loc_ws, O_ws);
    // 6) out = O @ W_out^T + b_out
    gemm_wmma_f32<<<gemm_blocks(LQ, DM), 256, 0, stream>>>(
        O_ws, W_out, b_out, out, LQ, DM, DM, 0);
}